// MultiHeadedAttention_57904749085042
// MI455X (gfx1250) — compile-verified
//
#include <hip/hip_runtime.h>
#include <hip/hip_bf16.h>
#include <cstdint>

#define B_  8
#define S_  1024
#define D_  1024
#define H_  16
#define HD_ 64

typedef __attribute__((ext_vector_type(2))) float v2f;
typedef __attribute__((ext_vector_type(8))) float v8f;

__device__ __forceinline__ v8f wmma_f32_k4(v2f a, v2f b, v8f c) {
  // D = A(16x4,f32) * B(4x16,f32) + C(16x16,f32)
  return __builtin_amdgcn_wmma_f32_16x16x4_f32(
      /*neg_a=*/false, a, /*neg_b=*/false, b,
      /*c_mod=*/(short)0, c, /*reuse_a=*/false, /*reuse_b=*/false);
}

// ---------------------------------------------------------------------------
// Kernel 1: fused projections, LDS-blocked GEMM.
//   which=0: K = key   @ Wk^T + bk
//   which=1: V = value @ Wv^T + bv
//   which=2: Q = (query @ Wv^T + bv) * 0.125  (faithful to reference bug+scale)
// 256 threads = 8 waves compute a 128x128 output tile; A/B staged through LDS
// in 128x32 chunks (36KB) -> 8x reuse of both operands vs. wave-private tiles.
// Wave grid 4(M) x 2(N); each wave owns a 32x64 accumulator (8 x v8f).
// Output layout: [B, H, S, hd] so attention reads are contiguous.
// ---------------------------------------------------------------------------
__global__ __launch_bounds__(256)
void qkv_proj_kernel(const float* __restrict__ key, const float* __restrict__ value,
                     const float* __restrict__ query,
                     const float* __restrict__ Wk, const float* __restrict__ bk,
                     const float* __restrict__ Wv, const float* __restrict__ bv,
                     float* __restrict__ Qh, float* __restrict__ Kh,
                     float* __restrict__ Vh) {
  __shared__ float As[128][36];   // 128 rows x 32 k, pad->36 (bank step 4 mod 64)
  __shared__ float Bs[128][36];

  const int which = blockIdx.z;
  const float* X    = (which == 0) ? key : ((which == 1) ? value : query);
  const float* W    = (which == 0) ? Wk  : Wv;
  const float* bias = (which == 0) ? bk  : bv;
  float*       outp = (which == 0) ? Kh  : ((which == 1) ? Vh : Qh);

  const int tid  = threadIdx.x;
  const int wave = tid >> 5;
  const int lane = tid & 31;
  const int l15  = lane & 15;
  const int koff = (lane >> 4) * 2;    // WMMA A/B layout: lanes 16..31 hold K=2,3
  const int wm   = wave >> 1;          // 0..3  (M wave index)
  const int wn   = wave & 1;           // 0..1  (N wave index)

  const int n0 = blockIdx.x * 128;
  const int m0 = blockIdx.y * 128;

  // Staging: 2 threads per row, 16 consecutive k each (4 x float4).
  const int srow = tid >> 1;
  const int scol = (tid & 1) * 16;
  const float* ag = X + (size_t)(m0 + srow) * D_ + scol;
  const float* bg = W + (size_t)(n0 + srow) * D_ + scol;

  v8f acc[2][4];
#pragma unroll
  for (int i = 0; i < 2; ++i)
#pragma unroll
    for (int j = 0; j < 4; ++j)
      acc[i][j] = (v8f){0.f, 0.f, 0.f, 0.f, 0.f, 0.f, 0.f, 0.f};

  for (int kc = 0; kc < D_; kc += 32) {
    __syncthreads();
    __builtin_prefetch(ag + kc + 32, 0, 3);   // global_prefetch_b8 (WGP-scope)
    __builtin_prefetch(bg + kc + 32, 0, 3);
#pragma unroll
    for (int q = 0; q < 16; q += 4) {
      *(float4*)&As[srow][scol + q] = *(const float4*)(ag + kc + q);
      *(float4*)&Bs[srow][scol + q] = *(const float4*)(bg + kc + q);
    }
    __syncthreads();
#pragma unroll
    for (int kk = 0; kk < 32; kk += 4) {
      v2f a0 = *(const v2f*)&As[wm * 32 + l15][kk + koff];
      v2f a1 = *(const v2f*)&As[wm * 32 + 16 + l15][kk + koff];
      v2f b0 = *(const v2f*)&Bs[wn * 64 + l15][kk + koff];
      v2f b1 = *(const v2f*)&Bs[wn * 64 + 16 + l15][kk + koff];
      v2f b2 = *(const v2f*)&Bs[wn * 64 + 32 + l15][kk + koff];
      v2f b3 = *(const v2f*)&Bs[wn * 64 + 48 + l15][kk + koff];
      acc[0][0] = wmma_f32_k4(a0, b0, acc[0][0]);
      acc[0][1] = wmma_f32_k4(a0, b1, acc[0][1]);
      acc[0][2] = wmma_f32_k4(a0, b2, acc[0][2]);
      acc[0][3] = wmma_f32_k4(a0, b3, acc[0][3]);
      acc[1][0] = wmma_f32_k4(a1, b0, acc[1][0]);
      acc[1][1] = wmma_f32_k4(a1, b1, acc[1][1]);
      acc[1][2] = wmma_f32_k4(a1, b2, acc[1][2]);
      acc[1][3] = wmma_f32_k4(a1, b3, acc[1][3]);
    }
  }

  // C/D layout: VGPR r -> row (r + 8*(lane>>4)), col (lane&15).
  const float scale = (which == 2) ? 0.125f : 1.0f;   // 1/sqrt(64) folded into Q
  const int mb = m0 + wm * 32 + (lane >> 4) * 8;
#pragma unroll
  for (int j = 0; j < 4; ++j) {
    const int n  = n0 + wn * 64 + j * 16 + l15;
    const float bn = bias[n];
    const int h = n >> 6, dh = n & 63;
#pragma unroll
    for (int i = 0; i < 2; ++i) {
#pragma unroll
      for (int r = 0; r < 8; ++r) {
        const int m = mb + i * 16 + r;
        const int b = m >> 10, s = m & 1023;
        outp[((size_t)(b * H_ + h) * S_ + s) * HD_ + dh] = (acc[i][j][r] + bn) * scale;
      }
    }
  }
}

// ---------------------------------------------------------------------------
// Kernel 2: attention for one (b, h, 16-query-row) strip.
// 256 threads = 8 waves. Q tile (4KB) and the full 16x1024 score block (64KB)
// live in LDS (320KB/WGP). WMMA scores -> masked softmax in LDS (attn written
// once to HBM) -> WMMA context = attn @ V with attn A-operand read from LDS.
// ---------------------------------------------------------------------------
__global__ __launch_bounds__(256)
void attn_kernel(const float* __restrict__ Qh, const float* __restrict__ Kh,
                 const float* __restrict__ Vh, const uint8_t* __restrict__ mask,
                 float* __restrict__ attn_out, float* __restrict__ Ctx) {
  __shared__ float sc[16][1028];     // 16x1024 scores, +4 pad to spread 64 banks
  __shared__ float qs[16][68];       // staged Q tile (read 64x otherwise)
  __shared__ float red[16][16];      // per-row partial reductions
  __shared__ float rowstat[16][2];   // row max, 1/rowsum
  __shared__ float ctxred[4][256];   // cross-wave K-split reduction

  const int q0   = blockIdx.x * 16;
  const int h    = blockIdx.y;
  const int b    = blockIdx.z;
  const int tid  = threadIdx.x;
  const int wave = tid >> 5;
  const int lane = tid & 31;
  const int l15  = lane & 15;
  const int koff = (lane >> 4) * 2;

  const size_t bh = (size_t)(b * H_ + h);
  const float* qb = Qh + (bh * S_ + q0) * HD_;
  const float* kb = Kh + bh * S_ * HD_;
  const float* vb = Vh + bh * S_ * HD_;

  // ---- Stage Q tile: 16 rows x 64, one float4 per thread.
  {
    const int r  = tid >> 4;
    const int c4 = (tid & 15) * 4;
    *(float4*)&qs[r][c4] = *(const float4*)(qb + (size_t)r * HD_ + c4);
  }
  __syncthreads();

  // ---- Phase 1: scores = Q * K^T (Q pre-scaled). 64 col-chunks over 8 waves.
  for (int chunk = wave; chunk < 64; chunk += 8) {
    const int c0 = chunk * 16;
    const float* bp = kb + (size_t)(c0 + l15) * HD_ + koff;
    v8f acc = {0.f, 0.f, 0.f, 0.f, 0.f, 0.f, 0.f, 0.f};
#pragma unroll
    for (int k0 = 0; k0 < HD_; k0 += 4) {
      v2f a  = *(const v2f*)&qs[l15][k0 + koff];
      v2f bb = *(const v2f*)(bp + k0);
      acc = wmma_f32_k4(a, bb, acc);
    }
    const int rbase = (lane >> 4) * 8;
#pragma unroll
    for (int r = 0; r < 8; ++r) sc[rbase + r][c0 + l15] = acc[r];
  }
  __syncthreads();

  // ---- Phase 2: mask + softmax, 16 threads per row.
  const int row = tid >> 4;   // 0..15
  const int t16 = tid & 15;   // 0..15
  const uint8_t* mrow = mask + ((size_t)b * S_ + (q0 + row)) * S_;

  float mx = -3.0e38f;
  for (int c = t16; c < S_; c += 16) {
    float v = mrow[c] ? -1.0e18f : sc[row][c];
    sc[row][c] = v;
    mx = fmaxf(mx, v);
  }
  red[row][t16] = mx;
  __syncthreads();
  if (t16 == 0) {
    float m2 = red[row][0];
#pragma unroll
    for (int i = 1; i < 16; ++i) m2 = fmaxf(m2, red[row][i]);
    rowstat[row][0] = m2;
  }
  __syncthreads();
  const float rmax = rowstat[row][0];
  float psum = 0.f;
  for (int c = t16; c < S_; c += 16) {
    float p = __expf(sc[row][c] - rmax);
    sc[row][c] = p;
    psum += p;
  }
  red[row][t16] = psum;
  __syncthreads();
  if (t16 == 0) {
    float s2 = 0.f;
#pragma unroll
    for (int i = 0; i < 16; ++i) s2 += red[row][i];
    rowstat[row][1] = 1.0f / s2;
  }
  __syncthreads();
  const float rinv = rowstat[row][1];
  float* aout = attn_out + (bh * S_ + (q0 + row)) * S_;
  for (int c = t16; c < S_; c += 16) {
    float a = sc[row][c] * rinv;
    sc[row][c] = a;       // kept in LDS for the context GEMM
    aout[c]    = a;       // attn output written exactly once
  }
  __syncthreads();

  // ---- Phase 3: context[16x64] = attn[16x1024] @ V[1024x64].
  // 4 N-tiles x 2 K-halves across the 8 waves, reduce halves through LDS.
  const int tile = wave & 3;
  const int half = wave >> 2;
  const int n0   = tile * 16;
  const int kbeg = half * 512;
  v8f cacc = {0.f, 0.f, 0.f, 0.f, 0.f, 0.f, 0.f, 0.f};
  for (int k0 = kbeg; k0 < kbeg + 512; k0 += 4) {
    v2f a;
    a.x = sc[l15][k0 + koff];
    a.y = sc[l15][k0 + koff + 1];
    const float* vp = vb + (size_t)(k0 + koff) * HD_ + n0 + l15;
    v2f bb;
    bb.x = vp[0];
    bb.y = vp[HD_];
    cacc = wmma_f32_k4(a, bb, cacc);
  }
  if (half == 1) {
#pragma unroll
    for (int r = 0; r < 8; ++r) ctxred[tile][lane + 32 * r] = cacc[r];
  }
  __syncthreads();
  if (half == 0) {
    float* cb = Ctx + (bh * S_ + q0) * HD_;
    const int rbase = (lane >> 4) * 8;
#pragma unroll
    for (int r = 0; r < 8; ++r) {
      float v = cacc[r] + ctxred[tile][lane + 32 * r];
      cb[(size_t)(rbase + r) * HD_ + n0 + l15] = v;
    }
  }
}

// ---------------------------------------------------------------------------
// Kernel 3: output = context @ Wo^T + bo. Same LDS-blocked GEMM as kernel 1;
// the A staging un-permutes [B,H,S,hd] -> [B,S,D] while filling LDS.
// ---------------------------------------------------------------------------
__global__ __launch_bounds__(256)
void out_proj_kernel(const float* __restrict__ Ctx, const float* __restrict__ Wo,
                     const float* __restrict__ bo, float* __restrict__ out) {
  __shared__ float As[128][36];
  __shared__ float Bs[128][36];

  const int tid  = threadIdx.x;
  const int wave = tid >> 5;
  const int lane = tid & 31;
  const int l15  = lane & 15;
  const int koff = (lane >> 4) * 2;
  const int wm   = wave >> 1;
  const int wn   = wave & 1;

  const int n0 = blockIdx.x * 128;
  const int m0 = blockIdx.y * 128;

  const int srow = tid >> 1;
  const int scol = (tid & 1) * 16;
  const int sm = m0 + srow;
  const int sb = sm >> 10, ss = sm & 1023;
  const float* abase = Ctx + ((size_t)sb * H_ * S_ + ss) * HD_;  // + hh*S*hd + dh
  const float* bg    = Wo + (size_t)(n0 + srow) * D_ + scol;

  v8f acc[2][4];
#pragma unroll
  for (int i = 0; i < 2; ++i)
#pragma unroll
    for (int j = 0; j < 4; ++j)
      acc[i][j] = (v8f){0.f, 0.f, 0.f, 0.f, 0.f, 0.f, 0.f, 0.f};

  for (int kc = 0; kc < D_; kc += 32) {
    __syncthreads();
    __builtin_prefetch(bg + kc + 32, 0, 3);
#pragma unroll
    for (int q = 0; q < 16; q += 4) {
      const int k  = kc + scol + q;         // 4-aligned; float4 stays in one head
      const int hh = k >> 6;
      const int dh = k & 63;
      *(float4*)&As[srow][scol + q] = *(const float4*)(abase + (size_t)hh * S_ * HD_ + dh);
      *(float4*)&Bs[srow][scol + q] = *(const float4*)(bg + kc + q);
    }
    __syncthreads();
#pragma unroll
    for (int kk = 0; kk < 32; kk += 4) {
      v2f a0 = *(const v2f*)&As[wm * 32 + l15][kk + koff];
      v2f a1 = *(const v2f*)&As[wm * 32 + 16 + l15][kk + koff];
      v2f b0 = *(const v2f*)&Bs[wn * 64 + l15][kk + koff];
      v2f b1 = *(const v2f*)&Bs[wn * 64 + 16 + l15][kk + koff];
      v2f b2 = *(const v2f*)&Bs[wn * 64 + 32 + l15][kk + koff];
      v2f b3 = *(const v2f*)&Bs[wn * 64 + 48 + l15][kk + koff];
      acc[0][0] = wmma_f32_k4(a0, b0, acc[0][0]);
      acc[0][1] = wmma_f32_k4(a0, b1, acc[0][1]);
      acc[0][2] = wmma_f32_k4(a0, b2, acc[0][2]);
      acc[0][3] = wmma_f32_k4(a0, b3, acc[0][3]);
      acc[1][0] = wmma_f32_k4(a1, b0, acc[1][0]);
      acc[1][1] = wmma_f32_k4(a1, b1, acc[1][1]);
      acc[1][2] = wmma_f32_k4(a1, b2, acc[1][2]);
      acc[1][3] = wmma_f32_k4(a1, b3, acc[1][3]);
    }
  }

  const int mb = m0 + wm * 32 + (lane >> 4) * 8;
#pragma unroll
  for (int j = 0; j < 4; ++j) {
    const int n  = n0 + wn * 64 + j * 16 + l15;
    const float bn = bo[n];
#pragma unroll
    for (int i = 0; i < 2; ++i) {
#pragma unroll
      for (int r = 0; r < 8; ++r) {
        out[(size_t)(mb + i * 16 + r) * D_ + n] = acc[i][j][r] + bn;
      }
    }
  }
}

// ---------------------------------------------------------------------------
extern "C" void kernel_launch(void* const* d_in, const int* in_sizes, int n_in,
                              void* d_out, int out_size, void* d_ws, size_t ws_size,
                              hipStream_t stream) {
  (void)in_sizes; (void)n_in; (void)out_size; (void)ws_size;

  const float*   key   = (const float*)d_in[0];
  const float*   value = (const float*)d_in[1];
  const float*   query = (const float*)d_in[2];
  const uint8_t* mask  = (const uint8_t*)d_in[3];
  const float*   Wk    = (const float*)d_in[4];
  const float*   bk    = (const float*)d_in[5];
  const float*   Wv    = (const float*)d_in[6];
  const float*   bv    = (const float*)d_in[7];
  const float*   Wo    = (const float*)d_in[8];
  const float*   bo    = (const float*)d_in[9];

  float* out      = (float*)d_out;
  float* attn_out = out + (size_t)B_ * S_ * D_;   // outputs concatenated: (output, attn)

  const size_t N1 = (size_t)B_ * H_ * S_ * HD_;   // 8,388,608 floats per tensor
  float* ws  = (float*)d_ws;
  float* Qh  = ws;
  float* Kh  = ws + N1;
  float* Vh  = ws + 2 * N1;
  float* Ctx = ws + 3 * N1;

  qkv_proj_kernel<<<dim3(D_ / 128, (B_ * S_) / 128, 3), 256, 0, stream>>>(
      key, value, query, Wk, bk, Wv, bv, Qh, Kh, Vh);

  attn_kernel<<<dim3(S_ / 16, H_, B_), 256, 0, stream>>>(
      Qh, Kh, Vh, mask, attn_out, Ctx);

  out_proj_kernel<<<dim3(D_ / 128, (B_ * S_) / 128), 256, 0, stream>>>(
      Ctx, Wo, bo, out);
}